// MultiheadAttPoolLayer_377957122130
// MI455X (gfx1250) — compile-verified
//
#include <hip/hip_runtime.h>
#include <math.h>

// Shapes
#define NH 8
#define DKH 128
#define DD 1024
#define BB 64
#define LL 2048

typedef __attribute__((ext_vector_type(2))) float v2f;
typedef __attribute__((ext_vector_type(8))) float v8f;

static __device__ __forceinline__ v8f wmma_f32(v2f a, v2f b, v8f c) {
  // V_WMMA_F32_16X16X4_F32 : D = A(16x4) * B(4x16) + C(16x16)
  return __builtin_amdgcn_wmma_f32_16x16x4_f32(false, a, false, b, (short)0, c,
                                               false, false);
}

// ---------------------------------------------------------------------------
// Kernel 1: qs[b, r] = sum_j q[b,j] * Wq[r,j] + bq[r]
// WMMA: A = Wq rows (M=r), B = q rows (N=b), K = j. Full 16x16 utilization.
// grid 32 x 256  (256 waves: 64 r-tiles x 4 b-tiles)
// ---------------------------------------------------------------------------
__global__ void k_qs(const float* __restrict__ q, const float* __restrict__ wq,
                     const float* __restrict__ bq, float* __restrict__ qs) {
  int w = blockIdx.x * 8 + (threadIdx.x >> 5);
  int lane = threadIdx.x & 31;
  int m16 = lane & 15, hi = lane >> 4;
  int r0 = (w >> 2) * 16, b0 = (w & 3) * 16;
  const float* ap = wq + (size_t)(r0 + m16) * DD + 2 * hi;
  const float* bp = q + (size_t)(b0 + m16) * DD + 2 * hi;
  v8f acc = {};
#pragma unroll 8
  for (int j = 0; j < DD; j += 4) {
    v2f a = *(const v2f*)(ap + j);
    v2f b = *(const v2f*)(bp + j);
    acc = wmma_f32(a, b, acc);
  }
  int bb = b0 + m16;
#pragma unroll
  for (int v = 0; v < 8; ++v) {
    int r = r0 + v + 8 * hi;
    qs[(size_t)bb * DD + r] = acc[v] + bq[r];
  }
}

// ---------------------------------------------------------------------------
// Kernel 2: qt[b,n,j] = sum_d qs[b, n*128+d] * Wk[n*128+d, j]
// WMMA: A = qs rows (M=b), B = Wk (K=d, N=j).
// grid 256 x 256 (2048 waves: n(8) x b-tile(4) x j-tile(64))
// ---------------------------------------------------------------------------
__global__ void k_qt(const float* __restrict__ qs, const float* __restrict__ wk,
                     float* __restrict__ qt) {
  int w = blockIdx.x * 8 + (threadIdx.x >> 5);
  int lane = threadIdx.x & 31;
  int m16 = lane & 15, hi = lane >> 4;
  int n = w >> 8;
  int rem = w & 255;
  int b0 = (rem >> 6) * 16, j0 = (rem & 63) * 16;
  const float* ap = qs + (size_t)(b0 + m16) * DD + n * DKH + 2 * hi;
  const float* bp = wk + (size_t)(n * DKH + 2 * hi) * DD + j0 + m16;
  v8f acc = {};
#pragma unroll 4
  for (int d = 0; d < DKH; d += 4) {
    v2f a = *(const v2f*)(ap + d);
    v2f b;
    b.x = bp[(size_t)d * DD];
    b.y = bp[(size_t)(d + 1) * DD];
    acc = wmma_f32(a, b, acc);
  }
#pragma unroll
  for (int v = 0; v < 8; ++v) {
    int bb = b0 + v + 8 * hi;
    qt[((size_t)bb * NH + n) * DD + j0 + m16] = acc[v];
  }
}

// ---------------------------------------------------------------------------
// Kernel 3: c[b,n] = sum_d qs[b, n*128+d] * bk[n*128+d]   (512 dots, tiny)
// ---------------------------------------------------------------------------
__global__ void k_c(const float* __restrict__ qs, const float* __restrict__ bk,
                    float* __restrict__ c) {
  int t = blockIdx.x * blockDim.x + threadIdx.x;
  if (t >= BB * NH) return;
  int b = t >> 3, n = t & 7;
  float s = 0.f;
  for (int d = 0; d < DKH; ++d)
    s += qs[(size_t)b * DD + n * DKH + d] * bk[n * DKH + d];
  c[t] = s;
}

// ---------------------------------------------------------------------------
// Kernel 4 (streaming pass 1 over k, 512 MB, read exactly once):
//   logits[b,n,l] = (qt[b,n,:] . k[b,l,:] + c[b,n]) / sqrt(128), masked
// WMMA: A = qt (M=n, padded to 16, from LDS), B = k rows (N=l, contiguous b64).
// LDS row stride 1028 floats -> conflict-free b64 ds loads.
// grid (64 b, 16 l-blocks) x 256; each wave = one 16-l tile.
// ---------------------------------------------------------------------------
__global__ void k_logits(const float* __restrict__ k,
                         const float* __restrict__ qt,
                         const float* __restrict__ cws,
                         const unsigned char* __restrict__ mask,
                         float* __restrict__ logits) {
  __shared__ float at[16 * 1028];
  int b = blockIdx.x;
  int tid = threadIdx.x;
  for (int i = tid; i < 16 * 1024; i += 256) {
    int row = i >> 10, col = i & 1023;
    at[row * 1028 + col] =
        (row < NH) ? qt[((size_t)b * NH + row) * DD + col] : 0.f;
  }
  __syncthreads();
  int wv = tid >> 5, lane = tid & 31, m16 = lane & 15, hi = lane >> 4;
  int l0 = blockIdx.y * 128 + wv * 16;
  const float* bp = k + ((size_t)b * LL + l0 + m16) * DD + 2 * hi;
  const float* ap = at + m16 * 1028 + 2 * hi;
  v8f acc = {};
#pragma unroll 4
  for (int j = 0; j < DD; j += 4) {
    v2f a = *(const v2f*)(ap + j);
    v2f bv = *(const v2f*)(bp + j);
    acc = wmma_f32(a, bv, acc);
  }
  const float invT = 0.08838834764831845f;  // 1/sqrt(128)
  if (hi == 0) {
    int l = l0 + m16;
    bool msk = mask[(size_t)b * LL + l] != 0;
#pragma unroll
    for (int v = 0; v < 8; ++v) {
      float val = (acc[v] + cws[b * NH + v]) * invT;
      logits[((size_t)b * NH + v) * LL + l] = msk ? -__builtin_inff() : val;
    }
  }
}

// ---------------------------------------------------------------------------
// Kernel 5: row softmax; writes attn section of d_out, head-major (n*B+b, l).
// Each thread owns 8 elements -> row cached in registers: one read, one exp,
// one write per element.
// grid 512 (= b*8+n) x 256
// ---------------------------------------------------------------------------
__global__ void k_softmax(const float* __restrict__ logits,
                          float* __restrict__ attn_out) {
  __shared__ float red[256];
  int row = blockIdx.x;
  int b = row >> 3, n = row & 7;
  const float* src = logits + (size_t)row * LL;
  float* dst = attn_out + (size_t)(n * BB + b) * LL;
  int tid = threadIdx.x;
  float x[8];
  float m = -__builtin_inff();
#pragma unroll
  for (int j = 0; j < 8; ++j) {
    x[j] = src[tid + 256 * j];
    m = fmaxf(m, x[j]);
  }
  red[tid] = m;
  __syncthreads();
  for (int s = 128; s > 0; s >>= 1) {
    if (tid < s) red[tid] = fmaxf(red[tid], red[tid + s]);
    __syncthreads();
  }
  m = red[0];
  __syncthreads();
  float e[8];
  float sum = 0.f;
#pragma unroll
  for (int j = 0; j < 8; ++j) {
    e[j] = __expf(x[j] - m);
    sum += e[j];
  }
  red[tid] = sum;
  __syncthreads();
  for (int s = 128; s > 0; s >>= 1) {
    if (tid < s) red[tid] += red[tid + s];
    __syncthreads();
  }
  float inv = 1.0f / red[0];
#pragma unroll
  for (int j = 0; j < 8; ++j) dst[tid + 256 * j] = e[j] * inv;
}

// ---------------------------------------------------------------------------
// Kernel 6 (streaming pass 2 over k, 512 MB, read exactly once):
//   s[b,n,d] = sum_l attn[b,n,l] * k[b,l,d]
// WMMA: A = attn (M=n from LDS), B = k (K=l, N=d).
// L processed in two 1024-wide halves so LDS is ~64 KB (stride 1028,
// conflict-free) -> 4 workgroups per WGP instead of 2.
// grid (64 b, 8 d-blocks) x 256
// ---------------------------------------------------------------------------
__global__ void k_s(const float* __restrict__ k,
                    const float* __restrict__ attn, float* __restrict__ sws) {
  __shared__ float at[16 * 1028];
  int b = blockIdx.x;
  int tid = threadIdx.x;
  int wv = tid >> 5, lane = tid & 31, m16 = lane & 15, hi = lane >> 4;
  int d0 = blockIdx.y * 128 + wv * 16;
  const float* bpbase = k + ((size_t)b * LL + 2 * hi) * DD + d0 + m16;
  v8f acc = {};
#pragma unroll 1
  for (int half = 0; half < 2; ++half) {
    __syncthreads();  // previous half fully consumed before overwrite
    for (int i = tid; i < 16 * 1024; i += 256) {
      int row = i >> 10, col = i & 1023;
      at[row * 1028 + col] =
          (row < NH)
              ? attn[((size_t)(row * BB + b)) * LL + half * 1024 + col]
              : 0.f;
    }
    __syncthreads();
    const float* ap = at + m16 * 1028 + 2 * hi;
    const float* bp = bpbase + (size_t)half * 1024 * DD;
#pragma unroll 4
    for (int l = 0; l < 1024; l += 4) {
      v2f a = *(const v2f*)(ap + l);
      v2f bv;
      bv.x = bp[(size_t)l * DD];
      bv.y = bp[(size_t)(l + 1) * DD];
      acc = wmma_f32(a, bv, acc);
    }
  }
  if (hi == 0) {
#pragma unroll
    for (int v = 0; v < 8; ++v)
      sws[((size_t)b * NH + v) * DD + d0 + m16] = acc[v];
  }
}

// ---------------------------------------------------------------------------
// Kernel 7: out[b, n*128+d] = sum_j Wv[n*128+d, j] * s[b,n,j] + bv[n*128+d]
// WMMA: A = Wv rows (M=d-row), B = s (K=j, N=b). Full 16x16 utilization.
// grid 32 x 256 (256 waves: n(8) x d-tile(8) x b-tile(4))
// ---------------------------------------------------------------------------
__global__ void k_out(const float* __restrict__ wvw,
                      const float* __restrict__ bv,
                      const float* __restrict__ sws, float* __restrict__ out) {
  int w = blockIdx.x * 8 + (threadIdx.x >> 5);
  int lane = threadIdx.x & 31;
  int m16 = lane & 15, hi = lane >> 4;
  int n = w >> 5;
  int rem = w & 31;
  int r0 = n * DKH + (rem >> 2) * 16;
  int b0 = (rem & 3) * 16;
  const float* ap = wvw + (size_t)(r0 + m16) * DD + 2 * hi;
  const float* bp = sws + ((size_t)(b0 + m16) * NH + n) * DD + 2 * hi;
  v8f acc = {};
#pragma unroll 8
  for (int j = 0; j < DD; j += 4) {
    v2f a = *(const v2f*)(ap + j);
    v2f b = *(const v2f*)(bp + j);
    acc = wmma_f32(a, b, acc);
  }
  int bb = b0 + m16;
#pragma unroll
  for (int v = 0; v < 8; ++v) {
    int r = r0 + v + 8 * hi;
    out[(size_t)bb * (NH * DKH) + r] = acc[v] + bv[r];
  }
}

// ---------------------------------------------------------------------------
extern "C" void kernel_launch(void* const* d_in, const int* in_sizes, int n_in,
                              void* d_out, int out_size, void* d_ws,
                              size_t ws_size, hipStream_t stream) {
  const float* q = (const float*)d_in[0];
  const float* k = (const float*)d_in[1];
  const unsigned char* mask = (const unsigned char*)d_in[2];
  const float* w_qs_w = (const float*)d_in[3];
  const float* w_qs_b = (const float*)d_in[4];
  const float* w_ks_w = (const float*)d_in[5];
  const float* w_ks_b = (const float*)d_in[6];
  const float* w_vs_w = (const float*)d_in[7];
  const float* w_vs_b = (const float*)d_in[8];

  float* outf = (float*)d_out;
  float* attn_sec = outf + (size_t)BB * NH * DKH;  // 65536

  float* wsf = (float*)d_ws;
  float* qs_ws = wsf;                 //   65536 floats
  float* qt_ws = qs_ws + 65536;       //  524288
  float* c_ws = qt_ws + 524288;       //     512
  float* logits_ws = c_ws + 512;      // 1048576
  float* s_ws = logits_ws + 1048576;  //  524288   (total ~8.65 MB)

  k_qs<<<32, 256, 0, stream>>>(q, w_qs_w, w_qs_b, qs_ws);
  k_qt<<<256, 256, 0, stream>>>(qs_ws, w_ks_w, qt_ws);
  k_c<<<2, 256, 0, stream>>>(qs_ws, w_ks_b, c_ws);
  k_logits<<<dim3(BB, 16), 256, 0, stream>>>(k, qt_ws, c_ws, mask, logits_ws);
  k_softmax<<<BB * NH, 256, 0, stream>>>(logits_ws, attn_sec);
  k_s<<<dim3(BB, 8), 256, 0, stream>>>(k, attn_sec, s_ws);
  k_out<<<32, 256, 0, stream>>>(w_vs_w, w_vs_b, s_ws, outf);
}